// PwGaANLayer_65549790872120
// MI455X (gfx1250) — compile-verified
//
#include <hip/hip_runtime.h>
#include <hip/hip_bf16.h>
#include <math.h>

// ---------------------------------------------------------------------------
// PwGaAN layer for MI455X (gfx1250, wave32).
// Memory-bound problem (~1+ GB traffic, ~7 GFLOP) -> keep everything fp32 and
// use V_WMMA_F32_16X16X4_F32 for both dense GEMMs (exact vs reference).
// ---------------------------------------------------------------------------

typedef float v2f __attribute__((ext_vector_type(2)));
typedef float v8f __attribute__((ext_vector_type(8)));

#define DIM   64
#define SLOPE 0.01f
#define ENC_NEGINF 0x007FFFFFu   // enc(-inf)

// order-preserving float <-> uint mapping for atomicMax-based segment max
__device__ __forceinline__ unsigned encf(float f) {
  unsigned b = __float_as_uint(f);
  return (b & 0x80000000u) ? ~b : (b | 0x80000000u);
}
__device__ __forceinline__ float decf(unsigned u) {
  unsigned b = (u & 0x80000000u) ? (u ^ 0x80000000u) : ~u;
  return __uint_as_float(b);
}

// f32 WMMA 16x16x4:  D(16x16) += A(16x4) * B(4x16), all fp32.
__device__ __forceinline__ v8f wmma_f32(v2f a, v2f b, v8f c) {
  return __builtin_amdgcn_wmma_f32_16x16x4_f32(false, a, false, b, (short)0, c,
                                               false, false);
}

// ---------------------------------------------------------------------------
// 0) init workspace
// ---------------------------------------------------------------------------
__global__ void init_kernel(float* h, float* sumFeat, unsigned* maxFeatEnc,
                            float* denom, float* cnt, unsigned* segMaxEnc,
                            int N) {
  long long g = (long long)blockIdx.x * blockDim.x + threadIdx.x;
  long long n64 = (long long)N * DIM;
  if (g < n64) {
    h[g] = 0.0f;
    sumFeat[g] = 0.0f;
    maxFeatEnc[g] = ENC_NEGINF;
  }
  if (g < N) {
    denom[g] = 0.0f;
    cnt[g] = 0.0f;
    segMaxEnc[g] = ENC_NEGINF;
  }
}

// ---------------------------------------------------------------------------
// 1) prep: al2 = Wa^T a_l, ar2 = Wa^T a_r  (64 threads)
// ---------------------------------------------------------------------------
__global__ void prep_kernel(const float* __restrict__ Wa,
                            const float* __restrict__ a_l,
                            const float* __restrict__ a_r,
                            float* al2, float* ar2) {
  int k = threadIdx.x;
  float sl = 0.0f, sr2 = 0.0f;
  for (int o = 0; o < DIM; ++o) {
    float w = Wa[o * DIM + k];
    sl += a_l[o] * w;
    sr2 += a_r[o] * w;
  }
  al2[k] = sl;
  ar2[k] = sr2;
}

// ---------------------------------------------------------------------------
// 2) node projection z = v @ Wa^T via WMMA f32 16x16x4.
//    One wave (32 threads) per 16-row tile; K=64 in 16 steps.
// ---------------------------------------------------------------------------
__global__ void proj_kernel(const float* __restrict__ v,
                            const float* __restrict__ Wa,
                            float* __restrict__ z, int N) {
  __shared__ float sv[16][66];  // stride 66: even (8B-aligned f2), no bank conflicts
  const int lane = threadIdx.x;
  const int base = blockIdx.x * 16;
  for (int x = lane; x < 16 * DIM; x += 32) {
    int r = x >> 6, c = x & 63;
    int row = base + r;
    if (row >= N) row = N - 1;
    sv[r][c] = v[(long long)row * DIM + c];
  }
  __syncthreads();
  const int l15 = lane & 15, hi = lane >> 4;
  for (int t = 0; t < 4; ++t) {
    v8f acc = {};
    for (int s = 0; s < 16; ++s) {
      int k0 = (s << 2) + (hi << 1);
      v2f a, b;
      a.x = sv[l15][k0];
      a.y = sv[l15][k0 + 1];
      // B[k][n] = Wa[n][k] : lane n = 16t+l15
      const float* wp = Wa + (t * 16 + l15) * DIM + k0;
      b.x = wp[0];
      b.y = wp[1];
      acc = wmma_f32(a, b, acc);
    }
    for (int j = 0; j < 8; ++j) {
      int m = base + j + 8 * hi;
      if (m < N) z[(long long)m * DIM + t * 16 + l15] = acc[j];
    }
  }
}

// ---------------------------------------------------------------------------
// 3) s_r[n] = v[n] . ar2
// ---------------------------------------------------------------------------
__global__ void sr_kernel(const float* __restrict__ v,
                          const float* __restrict__ ar2, float* sr, int N) {
  int n = blockIdx.x * blockDim.x + threadIdx.x;
  if (n >= N) return;
  const float* vp = v + (long long)n * DIM;
  float s = 0.0f;
  for (int k = 0; k < DIM; ++k) s += vp[k] * ar2[k];
  sr[n] = s;
}

// ---------------------------------------------------------------------------
// 4) edge pass 1: 128 edges/block, 16 edges/wave.
//    - stage pwFeat = pre_w * v[src] in LDS
//    - e = leaky_relu(pwFeat . al2 + sr[dst]) ; atomicMax seg max
//    - gateProj = pwFeat @ Wg^T via WMMA ; atomicMax per (dst,col)
//    - atomicAdd pwFeat into sumFeat[dst], cnt[dst]++
// ---------------------------------------------------------------------------
__global__ void edge1_kernel(const float* __restrict__ v,
                             const float* __restrict__ prew,
                             const int* __restrict__ src,
                             const int* __restrict__ dst,
                             const float* __restrict__ Wg,
                             const float* __restrict__ al2,
                             const float* __restrict__ sr,
                             unsigned* maxFeatEnc, float* sumFeat, float* cnt,
                             unsigned* segMaxEnc, float* e_out, int E) {
  __shared__ float s_pw[128][66];
  __shared__ int s_src[128];
  __shared__ int s_dst[128];
  __shared__ float s_prw[128];
  __shared__ unsigned char s_val[128];

  const int tid = threadIdx.x;
  const long long eb = (long long)blockIdx.x * 128;

  if (tid < 128) {
    long long gi = eb + tid;
    bool vld = gi < E;
    long long gic = vld ? gi : (long long)(E - 1);
    s_src[tid] = src[gic];
    s_dst[tid] = dst[gic];
    s_prw[tid] = prew[gic];
    s_val[tid] = vld ? 1 : 0;
  }
  __syncthreads();

  for (int x = tid; x < 128 * DIM; x += 256) {
    int r = x >> 6, c = x & 63;
    s_pw[r][c] = s_prw[r] * v[(long long)s_src[r] * DIM + c];
  }
  __syncthreads();

  const int lane = tid & 31, wave = tid >> 5;
  const int l15 = lane & 15, hi = lane >> 4;
  const int rb = wave * 16;

  // ---- edge attention score (lanes 0..15 of each wave, one edge each) ----
  if (hi == 0) {
    int r = rb + l15;
    float acc = 0.0f;
    for (int c = 0; c < DIM; ++c) acc += s_pw[r][c] * al2[c];
    if (s_val[r]) {
      int d = s_dst[r];
      float e = acc + sr[d];
      e = e > 0.0f ? e : SLOPE * e;
      e_out[eb + r] = e;
      atomicMax(&segMaxEnc[d], encf(e));
    }
  }
  // reconverged here: EXEC all-ones for WMMA below

  // ---- gateProj = pwFeat @ Wg^T, then scatter atomicMax into maxFeat ----
  for (int t = 0; t < 4; ++t) {
    v8f acc = {};
    for (int s = 0; s < 16; ++s) {
      int k0 = (s << 2) + (hi << 1);
      v2f a, b;
      a.x = s_pw[rb + l15][k0];
      a.y = s_pw[rb + l15][k0 + 1];
      const float* wp = Wg + (t * 16 + l15) * DIM + k0;
      b.x = wp[0];
      b.y = wp[1];
      acc = wmma_f32(a, b, acc);
    }
    int col = t * 16 + l15;
    for (int j = 0; j < 8; ++j) {
      int r = rb + j + 8 * hi;
      if (s_val[r]) {
        int d = s_dst[r];
        atomicMax(&maxFeatEnc[(long long)d * DIM + col], encf(acc[j]));
      }
    }
  }

  // ---- mean-feature sums + counts ----
  for (int x = tid; x < 128 * DIM; x += 256) {
    int r = x >> 6, c = x & 63;
    if (s_val[r]) atomicAdd(&sumFeat[(long long)s_dst[r] * DIM + c], s_pw[r][c]);
  }
  if (tid < 128 && s_val[tid]) atomicAdd(&cnt[s_dst[tid]], 1.0f);
}

// ---------------------------------------------------------------------------
// 5) edge pass 2: ex = exp(e - segmax[dst]); denom[dst] += ex
// ---------------------------------------------------------------------------
__global__ void edge2_kernel(const float* __restrict__ e_in,
                             const int* __restrict__ dst,
                             const unsigned* __restrict__ segMaxEnc,
                             float* denom, float* ex_out, int E) {
  int i = blockIdx.x * blockDim.x + threadIdx.x;
  if (i >= E) return;
  int d = dst[i];
  float m = decf(segMaxEnc[d]);  // finite: this edge contributed to it
  float ex = __expf(e_in[i] - m);
  ex_out[i] = ex;
  atomicAdd(&denom[d], ex);
}

// ---------------------------------------------------------------------------
// 6) edge pass 3: h[dst] += alpha * z[src]   (64 lanes per edge)
// ---------------------------------------------------------------------------
__global__ void edge3_kernel(const float* __restrict__ z,
                             const int* __restrict__ src,
                             const int* __restrict__ dst,
                             const float* __restrict__ ex,
                             const float* __restrict__ denom, float* h,
                             long long total) {
  long long g = (long long)blockIdx.x * blockDim.x + threadIdx.x;
  if (g >= total) return;
  int i = (int)(g >> 6);
  int k = (int)(g & 63);
  int d = dst[i];
  float alpha = ex[i] / fmaxf(denom[d], 1e-16f);
  atomicAdd(&h[(long long)d * DIM + k], alpha * z[(long long)src[i] * DIM + k]);
}

// ---------------------------------------------------------------------------
// 7) final per-node gate: out = sigmoid(v.gl + maxFeat.gm + meanFeat.gr) * h
//    64 threads (2 waves) per node.
// ---------------------------------------------------------------------------
__global__ void node_final_kernel(const float* __restrict__ v,
                                  const unsigned* __restrict__ maxFeatEnc,
                                  const float* __restrict__ sumFeat,
                                  const float* __restrict__ cnt,
                                  const float* __restrict__ gl,
                                  const float* __restrict__ gm,
                                  const float* __restrict__ gr,
                                  const float* __restrict__ h,
                                  float* __restrict__ out, int N) {
  __shared__ float red[64];
  __shared__ float s_gate;
  int n = blockIdx.x;
  int k = threadIdx.x;
  long long idx = (long long)n * DIM + k;
  float mf = decf(maxFeatEnc[idx]);
  if (mf < -3.0e38f) mf = 0.0f;  // segment_max of empty segment -> 0
  float c = cnt[n];
  float mean = sumFeat[idx] / fmaxf(c, 1.0f);
  red[k] = v[idx] * gl[k] + mf * gm[k] + mean * gr[k];
  __syncthreads();
  if (k == 0) {
    float s = 0.0f;
    for (int j = 0; j < 64; ++j) s += red[j];
    s_gate = 1.0f / (1.0f + __expf(-s));
  }
  __syncthreads();
  out[idx] = s_gate * h[idx];
}

// ---------------------------------------------------------------------------
// host launcher
// ---------------------------------------------------------------------------
extern "C" void kernel_launch(void* const* d_in, const int* in_sizes, int n_in,
                              void* d_out, int out_size, void* d_ws,
                              size_t ws_size, hipStream_t stream) {
  const float* v = (const float*)d_in[0];
  const float* pre_w = (const float*)d_in[1];
  const int* src = (const int*)d_in[2];
  const int* dst = (const int*)d_in[3];
  const float* Wa = (const float*)d_in[4];
  const float* a_l = (const float*)d_in[5];
  const float* a_r = (const float*)d_in[6];
  const float* Wg = (const float*)d_in[7];
  const float* gl = (const float*)d_in[8];
  const float* gm = (const float*)d_in[9];
  const float* gr = (const float*)d_in[10];
  float* out = (float*)d_out;

  const int N = in_sizes[0] / DIM;
  const int E = in_sizes[2];
  const long long N64 = (long long)N * DIM;

  // workspace layout (all 4-byte elements)
  float* ws = (float*)d_ws;
  float* z = ws;                      // N*64
  float* sr = z + N64;                // N
  float* al2 = sr + N;                // 64
  float* ar2 = al2 + 64;              // 64
  float* e_buf = ar2 + 64;            // E
  float* ex_buf = e_buf + E;          // E
  unsigned* segMaxEnc = (unsigned*)(ex_buf + E);   // N
  float* denom = (float*)(segMaxEnc + N);          // N
  float* cnt = denom + N;                          // N
  unsigned* maxFeatEnc = (unsigned*)(cnt + N);     // N*64
  float* sumFeat = (float*)(maxFeatEnc + N64);     // N*64
  float* h = sumFeat + N64;                        // N*64

  const int B = 256;

  // 0) init
  init_kernel<<<(int)((N64 + B - 1) / B), B, 0, stream>>>(
      h, sumFeat, maxFeatEnc, denom, cnt, segMaxEnc, N);
  // 1) prep small vectors
  prep_kernel<<<1, 64, 0, stream>>>(Wa, a_l, a_r, al2, ar2);
  // 2) z = v @ Wa^T  (WMMA)
  proj_kernel<<<(N + 15) / 16, 32, 0, stream>>>(v, Wa, z, N);
  // 3) s_r
  sr_kernel<<<(N + B - 1) / B, B, 0, stream>>>(v, ar2, sr, N);
  // 4) edge pass 1 (WMMA gateProj + scores + max/mean scatter)
  edge1_kernel<<<(E + 127) / 128, 256, 0, stream>>>(
      v, pre_w, src, dst, Wg, al2, sr, maxFeatEnc, sumFeat, cnt, segMaxEnc,
      e_buf, E);
  // 5) softmax denominators
  edge2_kernel<<<(E + B - 1) / B, B, 0, stream>>>(e_buf, dst, segMaxEnc, denom,
                                                  ex_buf, E);
  // 6) weighted aggregation
  long long total = (long long)E * DIM;
  edge3_kernel<<<(int)((total + B - 1) / B), B, 0, stream>>>(
      z, src, dst, ex_buf, denom, h, total);
  // 7) gate + output
  node_final_kernel<<<N, 64, 0, stream>>>(v, maxFeatEnc, sumFeat, cnt, gl, gm,
                                          gr, h, out, N);
}